// SparseConvTransitionBlock_21698174779791
// MI455X (gfx1250) — compile-verified
//
#include <hip/hip_runtime.h>

typedef __attribute__((ext_vector_type(2))) float v2f;
typedef __attribute__((ext_vector_type(8))) float v8f;

#define K_BINS 9
#define CH_IN  32
#define C_A    24
#define C_B    8
#define KC     (K_BINS * CH_IN)   // 288 = reduction length
#define SROW   292                // padded LDS row stride (bank-conflict-free A reads)

// ---------------------------------------------------------------------------
// Kernel 1: segment boundaries via binary search over the SORTED out-index.
// ---------------------------------------------------------------------------
__global__ void seg_bounds_kernel(const int* __restrict__ nbr_out,
                                  int* __restrict__ seg_start,
                                  int n_out, int n_edges) {
    int n = blockIdx.x * blockDim.x + threadIdx.x;
    if (n > n_out) return;
    int lo = 0, hi = n_edges;
    while (lo < hi) {                       // lower_bound(nbr_out, n)
        int mid = (lo + hi) >> 1;
        if (nbr_out[mid] < n) lo = mid + 1; else hi = mid;
    }
    seg_start[n] = lo;
}

// ---------------------------------------------------------------------------
// Kernel 2: pre-pack weights into padded [KC x 32] / [KC x 16] B-matrices.
// ---------------------------------------------------------------------------
__global__ void pack_weights_kernel(const float* __restrict__ W_a,
                                    const float* __restrict__ W_b,
                                    float* __restrict__ WaP,
                                    float* __restrict__ WbP) {
    int t = blockIdx.x * blockDim.x + threadIdx.x;
    if (t < KC * 32) {
        int kc = t >> 5, f = t & 31;
        WaP[t] = (f < C_A) ? W_a[kc * C_A + f] : 0.0f;
    }
    if (t < KC * 16) {
        int kc = t >> 4, f = t & 15;
        WbP[t] = (f < C_B) ? W_b[kc * C_B + f] : 0.0f;
    }
}

// ---------------------------------------------------------------------------
// Kernel 3: fused segment-sum (LDS) + WMMA f32 GEMM + epilogue.
//   128 threads = 4 wave32 per 16 output points.
//   Waves 0..3 each accumulate 4 output points into LDS Sa/Sb.
//   Waves 0,1: branch-a N-tiles (cols 0-15, 16-23). Wave 2: branch-b tile.
// ---------------------------------------------------------------------------
__global__ __launch_bounds__(128) void fused_sparse_conv_kernel(
    const float* __restrict__ feats,
    const float* __restrict__ importance,
    const float* __restrict__ b_a,
    const float* __restrict__ b_b,
    const int*   __restrict__ nbr_idx,
    const int*   __restrict__ nbr_k,
    const int*   __restrict__ seg_start,
    const float* __restrict__ WaP,   // [KC][32]
    const float* __restrict__ WbP,   // [KC][16]
    float*       __restrict__ out,   // [n_out*32] feats1 ++ [n_out] out_imp
    int n_out)
{
    __shared__ float sA[16 * SROW];
    __shared__ float sB[16 * SROW];
    __shared__ float sImp[16];

    const int lane  = threadIdx.x & 31;
    const int wv    = threadIdx.x >> 5;   // 0..3
    const int pbase = blockIdx.x * 16;

    // ---- zero LDS accumulators ----
    for (int i = threadIdx.x; i < 16 * SROW; i += 128) { sA[i] = 0.0f; sB[i] = 0.0f; }
    if (threadIdx.x < 16) sImp[threadIdx.x] = 0.0f;
    __syncthreads();

    // ---- edge accumulation: wave wv exclusively owns point slots 4*wv..4*wv+3
    for (int pp = 0; pp < 4; ++pp) {
        const int ps = wv * 4 + pp;
        const int p  = pbase + ps;
        if (p >= n_out) break;
        const int s = seg_start[p];
        const int e = seg_start[p + 1];
        float impsum = 0.0f;
        float* __restrict__ sa = &sA[ps * SROW];
        float* __restrict__ sb = &sB[ps * SROW];

        for (int base = s; base < e; base += 32) {
            int cnt = e - base; if (cnt > 32) cnt = 32;
            int   vi = 0, vk = 0; float vm = 0.0f;
            if (lane < cnt) {
                vi = nbr_idx[base + lane];
                vk = nbr_k[base + lane];
                vm = importance[vi];
            }
            for (int i = 0; i < cnt; ++i) {
                const int   bi = __shfl(vi, i, 32);
                const int   bk = __shfl(vk, i, 32);
                const float bm = __shfl(vm, i, 32);
                const float v  = feats[bi * CH_IN + lane];   // coalesced 128B gather
                atomicAdd(&sa[bk * CH_IN + lane], v);        // ds_add_f32 (no rtn)
                atomicAdd(&sb[bk * CH_IN + lane], v * bm);
                impsum += bm;                                // identical in all lanes
            }
        }
        if (lane == 0) {
            sImp[ps] = impsum;
            out[n_out * 32 + p] = impsum;                    // out_imp
        }
    }
    __syncthreads();

    // ---- GEMM phase: 16x16 output tile per wave via V_WMMA_F32_16X16X4_F32
    if (wv >= 3) return;

    const bool  branch_b = (wv == 2);
    const int   ntile    = (wv == 1) ? 1 : 0;
    const int   bcols    = branch_b ? 16 : 32;
    const float* __restrict__ Wp = branch_b ? WbP : WaP;
    const float* __restrict__ S  = branch_b ? sB  : sA;

    const int m     = lane & 15;            // A row / B col within tile
    const int khalf = (lane >> 4) << 1;     // 0 (lanes 0-15) or 2 (lanes 16-31)

    v8f c = {};
    for (int kk = 0; kk < KC; kk += 4) {
        v2f a, b;
        // A: S[m][kk+khalf .. +1]  (contiguous pair -> ds_load_b64, conflict-free)
        a.x = S[m * SROW + kk + khalf];
        a.y = S[m * SROW + kk + khalf + 1];
        // B: Wp[kk+khalf][ntile*16 + m], Wp[kk+khalf+1][...]
        const float* wp = Wp + (kk + khalf) * bcols + ntile * 16 + m;
        b.x = wp[0];
        b.y = wp[bcols];
        c = __builtin_amdgcn_wmma_f32_16x16x4_f32(
                /*neg_a=*/false, a, /*neg_b=*/false, b,
                /*c_mod=*/(short)0, c, /*reuse_a=*/false, /*reuse_b=*/false);
    }

    // ---- epilogue: C/D layout -> VGPR r: lanes0-15 M=r, lanes16-31 M=r+8
    const int rowoff = (lane >> 4) << 3;    // +8 for upper half-wave
    #pragma unroll
    for (int r = 0; r < 8; ++r) {
        const int row = r + rowoff;
        const int gp  = pbase + row;
        if (gp >= n_out) continue;
        const int col = (lane & 15) + ntile * 16;
        const float val = c[r];
        if (!branch_b) {
            if (col < C_A) {
                float o = val + b_a[col];
                out[gp * 32 + col] = o > 0.0f ? o : 0.0f;
            }
        } else {
            if (col < C_B) {
                const float imp = sImp[row];
                const float den = imp > 0.0f ? imp : 1.0f;
                float o = val / den + b_b[col];
                out[gp * 32 + C_A + col] = o > 0.0f ? o : 0.0f;
            }
        }
    }
}

// ---------------------------------------------------------------------------
extern "C" void kernel_launch(void* const* d_in, const int* in_sizes, int n_in,
                              void* d_out, int out_size, void* d_ws, size_t ws_size,
                              hipStream_t stream) {
    const float* feats      = (const float*)d_in[0];
    const float* importance = (const float*)d_in[1];
    const float* W_a        = (const float*)d_in[2];
    const float* b_a        = (const float*)d_in[3];
    const float* W_b        = (const float*)d_in[4];
    const float* b_b        = (const float*)d_in[5];
    const int*   nbr_idx    = (const int*)d_in[6];
    const int*   nbr_k      = (const int*)d_in[7];
    const int*   nbr_out    = (const int*)d_in[8];

    const int n_edges = in_sizes[6];
    const int n_out   = out_size / 33;     // N_OUT*32 feats + N_OUT imp

    // workspace layout
    char* ws = (char*)d_ws;
    size_t seg_bytes = ((size_t)(n_out + 1) * sizeof(int) + 255) & ~(size_t)255;
    int*   seg_start = (int*)ws;
    float* WaP       = (float*)(ws + seg_bytes);
    float* WbP       = (float*)(ws + seg_bytes + (size_t)KC * 32 * sizeof(float));
    (void)ws_size; (void)n_in;

    seg_bounds_kernel<<<(n_out + 1 + 255) / 256, 256, 0, stream>>>(
        nbr_out, seg_start, n_out, n_edges);

    pack_weights_kernel<<<(KC * 32 + 255) / 256, 256, 0, stream>>>(
        W_a, W_b, WaP, WbP);

    fused_sparse_conv_kernel<<<(n_out + 15) / 16, 128, 0, stream>>>(
        feats, importance, b_a, b_b, nbr_idx, nbr_k,
        seg_start, WaP, WbP, (float*)d_out, n_out);
}